// DeconvLayer_31782757990857
// MI455X (gfx1250) — compile-verified
//
#include <hip/hip_runtime.h>

typedef __attribute__((ext_vector_type(2))) float v2f;
typedef __attribute__((ext_vector_type(8))) float v8f;

#define N_COLS 4096
#define N_ROWS 4096
#define T_BLOCKS (N_COLS / 16) /* 256 */
#define XSTRIDE 20  /* floats; 80B rows -> 16B aligned, conflict-free ds reads */
#define VSTRIDE 18  /* floats; conflict-free D-store / B-load patterns */

// Block recurrence (16-col blocks, 16-row batches), both IIR passes fused:
//   V_t = T*X_t + S*V_{t-1}
//   Y_t = U1*X_t + U2*V_{t-1} + S2*Y_{t-1}   with U1 = T2*T, U2 = T2*S
// Each product: 4 chained V_WMMA_F32_16X16X4_F32; the five 4-WMMA chains are
// mutually independent, so the loop-carried path is 4 WMMAs + v8f add + one
// in-order DS store/load (D->B fragment relayout) per recurrence.
//
// Fragment layouts (ISA 7.12.2, wave32):
//   A 16x4 chunk : lane holds M=lane&15, K = vgpr + 2*(lane>>4)
//   B 4x16 chunk : lane holds N=lane&15, K = vgpr + 2*(lane>>4)
//   C/D 16x16    : lane holds N=lane&15, M = vgpr + 8*(lane>>4)

__global__ __launch_bounds__(32) void deconv_iir_wmma(
    const float* __restrict__ x, const float* __restrict__ hker,
    float* __restrict__ out)
{
  const int g    = blockIdx.x;        // 16-row group
  const int lane = threadIdx.x;       // 0..31 (wave32)
  const int half = lane >> 4;
  const int l16  = lane & 15;

  // slots: 0=T, 1=S, 2=U1, 3=U2, 4=S2, 5=T2 (scratch)
  __shared__ float mats[6][16][16];
  __shared__ __align__(16) float xbuf[2][16 * XSTRIDE];
  __shared__ float vbuf[16 * VSTRIDE];
  __shared__ float ybuf[16 * VSTRIDE];

  // ---- lane 0: derive block-recurrence matrices from taps h ----
  if (lane == 0) {
    float a0[16], a1[16];
    for (int j = 0; j < 16; ++j) { a0[j] = hker[j]; a1[j] = hker[15 - j]; }
    for (int p = 0; p < 2; ++p) {
      const float* a = p ? a1 : a0;
      const int ts = p ? 5 : 0;   // T slot (pass1 T2 -> scratch slot 5)
      const int ss = p ? 4 : 1;   // S slot
      // impulse response g[n] -> T[n][m] = g[n-m] (lower triangular Toeplitz)
      float gi[16];
      for (int n = 0; n < 16; ++n) {
        float acc = (n == 0) ? 1.0f : 0.0f;
        for (int j = 1; j <= n; ++j) acc += a[j - 1] * gi[n - j];
        gi[n] = acc;
      }
      for (int n = 0; n < 16; ++n)
        for (int m = 0; m < 16; ++m)
          mats[ts][n][m] = (n >= m) ? gi[n - m] : 0.0f;
      // state response: S[n][m] = d v_n / d p_m  (p = previous block outputs)
      for (int m = 0; m < 16; ++m) {
        float w[32];
        for (int i2 = 0; i2 < 16; ++i2) w[i2] = (i2 == m) ? 1.0f : 0.0f;
        for (int n = 0; n < 16; ++n) {
          float acc = 0.0f;
          for (int j = 1; j <= 16; ++j) acc += a[j - 1] * w[16 + n - j];
          w[16 + n] = acc;
          mats[ss][n][m] = acc;
        }
      }
    }
    // U1 = T2*T, U2 = T2*S
    for (int n = 0; n < 16; ++n)
      for (int m = 0; m < 16; ++m) {
        float u1 = 0.0f, u2 = 0.0f;
        for (int k = 0; k < 16; ++k) {
          const float t2 = mats[5][n][k];
          u1 += t2 * mats[0][k][m];
          u2 += t2 * mats[1][k][m];
        }
        mats[2][n][m] = u1;
        mats[3][n][m] = u2;
      }
  }
  __syncthreads();

  // ---- resident A-operand fragments for T, S, U1, U2, S2 ----
  v2f A[5][4];
  #pragma unroll
  for (int m = 0; m < 5; ++m) {
    #pragma unroll
    for (int kk = 0; kk < 4; ++kk) {
      const int k0 = 4 * kk + 2 * half;
      v2f t; t.x = mats[m][l16][k0]; t.y = mats[m][l16][k0 + 1];
      A[m][kk] = t;
    }
  }

  const float* xg = x + (size_t)g * 16 * N_COLS;

  // async global -> LDS stage of one 16x16 X tile (two b128 ops, 64B/row segs)
  const unsigned r4 = (unsigned)(lane >> 2);   // row 0..7
  const unsigned q4 = (unsigned)(lane & 3);    // 16B quarter of the 64B row seg
  #define ISSUE_ASYNC(tn)                                                      \
    {                                                                          \
      unsigned lds0 = (unsigned)(size_t)(&xbuf[(tn) & 1][0]) +                 \
                      r4 * (XSTRIDE * 4u) + q4 * 16u;                          \
      unsigned lds1 = lds0 + 8u * (XSTRIDE * 4u);                              \
      unsigned vof0 = r4 * (N_COLS * 4u) + (unsigned)(tn) * 64u + q4 * 16u;    \
      unsigned vof1 = vof0 + 8u * (N_COLS * 4u);                               \
      asm volatile("global_load_async_to_lds_b128 %0, %1, %2"                  \
                   :: "v"(lds0), "v"(vof0), "s"(xg) : "memory");               \
      asm volatile("global_load_async_to_lds_b128 %0, %1, %2"                  \
                   :: "v"(lds1), "v"(vof1), "s"(xg) : "memory");               \
    }

  // reference semantics: first 16 outputs of each row are zero
  #pragma unroll
  for (int i = 0; i < 8; ++i)
    out[(size_t)(g * 16 + l16) * N_COLS + i + 8 * half] = 0.0f;

  v2f Bv[4] = {};   // v_{t-1} in B layout (zero state entering block 1)
  v2f By[4] = {};   // y_{t-1} in B layout

  ISSUE_ASYNC(1);

  for (int t = 1; t < T_BLOCKS; ++t) {
    if (t < T_BLOCKS - 1) {
      ISSUE_ASYNC(t + 1);
      asm volatile("s_wait_asynccnt 0x2" ::: "memory"); // tile t resident
    } else {
      asm volatile("s_wait_asynccnt 0x0" ::: "memory");
    }

    // X_t B-fragments from LDS tile [row][time]
    const float* xb = &xbuf[t & 1][0];
    v2f Bx[4];
    #pragma unroll
    for (int kk = 0; kk < 4; ++kk) {
      const int k0 = 4 * kk + 2 * half;
      v2f b; b.x = xb[l16 * XSTRIDE + k0]; b.y = xb[l16 * XSTRIDE + k0 + 1];
      Bx[kk] = b;
    }

    // five independent 4-WMMA chains
    v8f vA = {}, vB = {}, yA = {}, yB = {}, yC = {};
    #pragma unroll
    for (int kk = 0; kk < 4; ++kk) {
      vA = __builtin_amdgcn_wmma_f32_16x16x4_f32(
          false, A[0][kk], false, Bx[kk], (short)0, vA, false, false); // T*X
      vB = __builtin_amdgcn_wmma_f32_16x16x4_f32(
          false, A[1][kk], false, Bv[kk], (short)0, vB, false, false); // S*Vp
      yA = __builtin_amdgcn_wmma_f32_16x16x4_f32(
          false, A[2][kk], false, Bx[kk], (short)0, yA, false, false); // U1*X
      yB = __builtin_amdgcn_wmma_f32_16x16x4_f32(
          false, A[3][kk], false, Bv[kk], (short)0, yB, false, false); // U2*Vp
      yC = __builtin_amdgcn_wmma_f32_16x16x4_f32(
          false, A[4][kk], false, By[kk], (short)0, yC, false, false); // S2*Yp
    }
    const v8f vAcc = vA + vB;
    const v8f yAcc = (yA + yB) + yC;

    // V: D-layout -> B-layout swap through LDS (in-order DS, single wave)
    #pragma unroll
    for (int i = 0; i < 8; ++i)
      vbuf[(i + 8 * half) * VSTRIDE + l16] = vAcc[i];
    __builtin_amdgcn_wave_barrier();
    #pragma unroll
    for (int kk = 0; kk < 4; ++kk) {
      const int k0 = 4 * kk + 2 * half;
      v2f b; b.x = vbuf[k0 * VSTRIDE + l16]; b.y = vbuf[(k0 + 1) * VSTRIDE + l16];
      Bv[kk] = b;
    }

    // Y: same relayout
    #pragma unroll
    for (int i = 0; i < 8; ++i)
      ybuf[(i + 8 * half) * VSTRIDE + l16] = yAcc[i];
    __builtin_amdgcn_wave_barrier();
    #pragma unroll
    for (int kk = 0; kk < 4; ++kk) {
      const int k0 = 4 * kk + 2 * half;
      v2f b; b.x = ybuf[k0 * VSTRIDE + l16]; b.y = ybuf[(k0 + 1) * VSTRIDE + l16];
      By[kk] = b;
    }

    // store Y_t straight from D fragments: row = l16, col = 16t + i + 8*half
    float* yrow = out + (size_t)(g * 16 + l16) * N_COLS + t * 16;
    #pragma unroll
    for (int i = 0; i < 8; ++i)
      yrow[i + 8 * half] = yAcc[i];
  }
  #undef ISSUE_ASYNC
}

extern "C" void kernel_launch(void* const* d_in, const int* in_sizes, int n_in,
                              void* d_out, int out_size, void* d_ws, size_t ws_size,
                              hipStream_t stream) {
  (void)in_sizes; (void)n_in; (void)out_size; (void)d_ws; (void)ws_size;
  const float* x = (const float*)d_in[0];   // [4096, 4096] fp32
  const float* h = (const float*)d_in[1];   // [1, 16] fp32
  float* y = (float*)d_out;                 // [4096, 4096] fp32
  dim3 grid(N_ROWS / 16);                   // 256 groups, 1 wave each
  dim3 block(32);
  hipLaunchKernelGGL(deconv_iir_wmma, grid, block, 0, stream, x, h, y);
}